// get_model_88003879895121
// MI455X (gfx1250) — compile-verified
//
#include <hip/hip_runtime.h>

#define B_ 256
#define N_ 8192
#define S1_ 54
#define K1_ 28
#define S2_ 22
#define K2_ 8
#define M1_ (B_ * S1_ * K1_)   // 387072
#define M2_ (B_ * S2_ * K2_)   // 45056
#define M3_ (B_ * S2_)         // 5632
#define EPS_ 1e-5f

typedef float v2f __attribute__((ext_vector_type(2)));
typedef float v8f __attribute__((ext_vector_type(8)));

// ---------------------------------------------------------------------------
// Farthest point sampling: one workgroup per batch, xyz resident in LDS
// (n*3*4 bytes <= 96KB of the 320KB WGP LDS), dist in registers.
// ---------------------------------------------------------------------------
__global__ void fps_kernel(const float* __restrict__ pts, int n, int s,
                           float* __restrict__ out)
{
  extern __shared__ float sm[];          // n*3 floats
  __shared__ float rv[256];
  __shared__ int   ri[256];
  const int b = blockIdx.x, t = threadIdx.x;
  const float* P = pts + (size_t)b * n * 3;
  for (int i = t; i < n * 3; i += 256) sm[i] = P[i];
  __syncthreads();
  const int cnt = (n + 255) >> 8;        // points per thread (<=32)
  float dist[32];
  for (int i = 0; i < cnt; ++i) dist[i] = 1e10f;
  int far = 0;
  for (int it = 0; it < s; ++it) {
    if (t == 0) {
      out[((size_t)b * s + it) * 3 + 0] = sm[far * 3 + 0];
      out[((size_t)b * s + it) * 3 + 1] = sm[far * 3 + 1];
      out[((size_t)b * s + it) * 3 + 2] = sm[far * 3 + 2];
    }
    const float cx = sm[far * 3], cy = sm[far * 3 + 1], cz = sm[far * 3 + 2];
    float bv = -1.0f; int bi = 0;
    for (int i = 0; i < cnt; ++i) {
      const int j = t + (i << 8);
      if (j < n) {
        const float dx = sm[j * 3] - cx, dy = sm[j * 3 + 1] - cy, dz = sm[j * 3 + 2] - cz;
        const float nd = fminf(dist[i], dx * dx + dy * dy + dz * dz);
        dist[i] = nd;
        if (nd > bv) { bv = nd; bi = j; }
      }
    }
    rv[t] = bv; ri[t] = bi;
    __syncthreads();
    for (int off = 128; off > 0; off >>= 1) {
      if (t < off) {
        const float ov = rv[t + off]; const int oi = ri[t + off];
        if (ov > rv[t] || (ov == rv[t] && oi < ri[t])) { rv[t] = ov; ri[t] = oi; }
      }
      __syncthreads();
    }
    far = ri[0];
    __syncthreads();
  }
}

// ---------------------------------------------------------------------------
// Ball query + grouping, one wave32 per (batch, center). Selection order ==
// ascending point index (matches reference sort/truncate), padding with the
// first hit. Writes grouped rows [xyz-center | feats | 0-pad] with a channel
// stride padded to a multiple of 4 so the GEMM K-loop needs no tail guards.
// ---------------------------------------------------------------------------
__device__ __forceinline__ void write_group_row(
    float* __restrict__ dst, const float* __restrict__ P,
    const float* __restrict__ F, int jj, float cx, float cy, float cz,
    int fdim, int cdimp, int lane)
{
  for (int c = lane; c < cdimp; c += 32) {
    float v;
    if (c == 0)            v = P[jj * 3 + 0] - cx;
    else if (c == 1)       v = P[jj * 3 + 1] - cy;
    else if (c == 2)       v = P[jj * 3 + 2] - cz;
    else if (c < 3 + fdim) v = F[(size_t)jj * fdim + (c - 3)];
    else                   v = 0.f;                       // K padding
    dst[c] = v;
  }
}

__global__ void ball_group_kernel(
    const float* __restrict__ pts, const float* __restrict__ centers,
    const float* __restrict__ feats, float* __restrict__ Xout,
    int n, int s, int ns, int fdim, int cdimp, float r2)
{
  const int wave = (int)(((size_t)blockIdx.x * blockDim.x + threadIdx.x) >> 5);
  const int lane = threadIdx.x & 31;
  if (wave >= B_ * s) return;
  const int b = wave / s, si = wave % s;
  const float* C = centers + ((size_t)b * s + si) * 3;
  const float cx = C[0], cy = C[1], cz = C[2];
  const float* P = pts + (size_t)b * n * 3;
  const float* F = feats ? feats + (size_t)b * n * fdim : nullptr;
  float* rowbase = Xout + ((size_t)b * s + si) * ns * cdimp;
  int cnt = 0, firstj = -1;
  for (int base = 0; base < n && cnt < ns; base += 32) {
    const int j = base + lane;
    bool pred = false;
    if (j < n) {
      const float dx = P[j * 3] - cx, dy = P[j * 3 + 1] - cy, dz = P[j * 3 + 2] - cz;
      pred = (dx * dx + dy * dy + dz * dz) <= r2;
    }
    unsigned mask = (unsigned)__ballot(pred);      // wave32: low 32 bits
    if (firstj < 0 && mask) firstj = base + __ffs((int)mask) - 1;
    while (mask && cnt < ns) {
      const int lb = __ffs((int)mask) - 1;
      mask &= mask - 1;
      write_group_row(rowbase + (size_t)cnt * cdimp, P, F, base + lb, cx, cy, cz,
                      fdim, cdimp, lane);
      ++cnt;
    }
  }
  if (firstj < 0) firstj = 0;
  for (; cnt < ns; ++cnt)
    write_group_row(rowbase + (size_t)cnt * cdimp, P, F, firstj, cx, cy, cz,
                    fdim, cdimp, lane);
}

// Zero-pad a (Cout, Cin) weight matrix to (Cout, Cinp), Cinp % 4 == 0.
__global__ void pack_w_kernel(const float* __restrict__ W, float* __restrict__ Wp,
                              int Cout, int Cin, int Cinp)
{
  const int i = blockIdx.x * blockDim.x + threadIdx.x;
  if (i >= Cout * Cinp) return;
  const int o = i / Cinp, k = i % Cinp;
  Wp[i] = (k < Cin) ? W[(size_t)o * Cin + k] : 0.f;
}

// ---------------------------------------------------------------------------
// fp32 WMMA GEMM: Y[M,Cout] = X[M,Cin] @ W[Cout,Cin]^T + b.
// Requirements (guaranteed by this model + padding): M%16==0, Cout%16==0,
// Cin%4==0 (X row stride and W row stride both == Cin). Branch-free K-loop:
// one contiguous b64 A-load feeds NT WMMAs into NT independent accumulators.
//   MODE 0: store Y + per-channel sum/sumsq atomics (batch-norm stats)
//   MODE 1: stats only
//   MODE 3: recompute + normalize + ReLU + max-pool (atomicMax, v>=0)
// ---------------------------------------------------------------------------
template <int MODE, int NT>
__global__ void gemm16(const float* __restrict__ X, const float* __restrict__ W,
                       const float* __restrict__ bias, float* __restrict__ Y,
                       float* __restrict__ stats, const float* __restrict__ gamma,
                       const float* __restrict__ beta, float* __restrict__ pooled,
                       int M, int Cin, int Cout, int nsample)
{
  const int lane = threadIdx.x & 31;
  const int wid  = (int)(((size_t)blockIdx.x * blockDim.x + threadIdx.x) >> 5);
  const int groupsN = (Cout >> 4) / NT;          // exact by construction
  if (wid >= (M >> 4) * groupsN) return;
  const int tm = wid / groupsN, tg = wid % groupsN;
  const int khalf = (lane >> 4) << 1;            // 0 or 2
  const float* Xr = X + (size_t)(tm * 16 + (lane & 15)) * Cin + khalf;
  const int col0 = tg * (16 * NT) + (lane & 15);
  const float* Wr[NT];
#pragma unroll
  for (int t = 0; t < NT; ++t)
    Wr[t] = W + (size_t)(col0 + 16 * t) * Cin + khalf;

  v8f acc[NT];
#pragma unroll
  for (int t = 0; t < NT; ++t) acc[t] = (v8f){0.f,0.f,0.f,0.f,0.f,0.f,0.f,0.f};

  for (int k0 = 0; k0 < Cin; k0 += 4) {
    const v2f a = *(const v2f*)(Xr + k0);        // global_load_b64
#pragma unroll
    for (int t = 0; t < NT; ++t) {
      const v2f bm = *(const v2f*)(Wr[t] + k0);  // global_load_b64
#if __has_builtin(__builtin_amdgcn_wmma_f32_16x16x4_f32)
      acc[t] = __builtin_amdgcn_wmma_f32_16x16x4_f32(false, a, false, bm,
                                                     (short)0, acc[t], false, false);
#else
      acc[t][0] += a.x * bm.x + a.y * bm.y;      // placeholder
#endif
    }
  }

  const int rbase = tm * 16 + ((lane >> 4) << 3);   // D VGPR r -> row r / r+8
  const float invM = 1.0f / (float)M;

#pragma unroll
  for (int t = 0; t < NT; ++t) {
    const int col = col0 + 16 * t;
    const float bv = bias[col];
    if (MODE == 3) {
      const float mean = stats[col] * invM;
      const float var  = stats[1024 + col] * invM - mean * mean;
      const float rstd = rsqrtf(var + EPS_);
      const float ga = gamma[col], be = beta[col];
#pragma unroll
      for (int r = 0; r < 8; ++r) {
        const int rr = rbase + r;
        const float xn = fmaxf(ga * (acc[t][r] + bv - mean) * rstd + be, 0.f);
        atomicMax((unsigned int*)&pooled[(size_t)(rr / nsample) * Cout + col],
                  __float_as_uint(xn));
      }
    } else {
      float s = 0.f, sq = 0.f;
#pragma unroll
      for (int r = 0; r < 8; ++r) {
        const float v = acc[t][r] + bv;
        if (MODE == 0) Y[(size_t)(rbase + r) * Cout + col] = v;
        s += v; sq += v * v;
      }
      s  += __shfl_xor(s, 16);   // partner lane has the same column
      sq += __shfl_xor(sq, 16);
      if (lane < 16) {
        atomicAdd(&stats[col], s);
        atomicAdd(&stats[1024 + col], sq);
      }
    }
  }
}

__global__ void bn_relu_kernel(float* __restrict__ Y, const float* __restrict__ stats,
                               const float* __restrict__ gamma,
                               const float* __restrict__ beta, int M, int C)
{
  const size_t i = (size_t)blockIdx.x * blockDim.x + threadIdx.x;
  if (i >= (size_t)M * C) return;
  const int c = (int)(i % C);
  const float invM = 1.0f / (float)M;
  const float mean = stats[c] * invM;
  const float var  = stats[1024 + c] * invM - mean * mean;
  const float v = gamma[c] * (Y[i] - mean) * rsqrtf(var + EPS_) + beta[c];
  Y[i] = fmaxf(v, 0.f);
}

__global__ void concat_x3_kernel(const float* __restrict__ l2xyz,
                                 const float* __restrict__ l2pts,
                                 float* __restrict__ X3, int rows)
{
  const size_t i = (size_t)blockIdx.x * blockDim.x + threadIdx.x;
  if (i >= (size_t)rows * 132) return;               // 131 padded to 132
  const int g = (int)(i / 132), c = (int)(i % 132);
  float v;
  if (c < 3)        v = l2xyz[(size_t)g * 3 + c];
  else if (c < 131) v = l2pts[(size_t)g * 128 + (c - 3)];
  else              v = 0.f;
  X3[i] = v;
}

__global__ void fc3_kernel(const float* __restrict__ X, const float* __restrict__ W,
                           const float* __restrict__ bias, float* __restrict__ L)
{
  const int i = blockIdx.x * blockDim.x + threadIdx.x;   // B_*40
  if (i >= B_ * 40) return;
  const int r = i / 40, c = i % 40;
  const float* x = X + (size_t)r * 160;
  const float* w = W + (size_t)c * 160;
  float s = bias[c];
  for (int k = 0; k < 160; ++k) s += x[k] * w[k];
  L[i] = s;
}

__global__ void lsm_kernel(const float* __restrict__ L, float* __restrict__ out)
{
  const int wave = (int)(((size_t)blockIdx.x * blockDim.x + threadIdx.x) >> 5);
  const int lane = threadIdx.x & 31;
  if (wave >= B_) return;
  const float* l = L + (size_t)wave * 40;
  const float v1 = l[lane];
  const float v2 = (lane < 8) ? l[32 + lane] : -1e30f;
  float m = fmaxf(v1, v2);
  for (int o = 16; o > 0; o >>= 1) m = fmaxf(m, __shfl_xor(m, o));
  float e = __expf(v1 - m) + ((lane < 8) ? __expf(v2 - m) : 0.f);
  for (int o = 16; o > 0; o >>= 1) e += __shfl_xor(e, o);
  const float ls = m + __logf(e);
  out[(size_t)wave * 40 + lane] = v1 - ls;
  if (lane < 8) out[(size_t)wave * 40 + 32 + lane] = v2 - ls;
}

// ---------------------------------------------------------------------------
extern "C" void kernel_launch(void* const* d_in, const int* in_sizes, int n_in,
                              void* d_out_v, int out_size, void* d_ws, size_t ws_size,
                              hipStream_t stream)
{
  (void)n_in; (void)ws_size;
  const float* data = (const float*)d_in[0];
  // Layer slots: 0-2 = sa1, 3-5 = sa2, 6-8 = sa3, 9 = fc1, 10 = fc2.
  const float *Wl[11], *bl[11], *gl[11], *bel[11];
  const float *fc3W, *fc3b;
  if (in_sizes[1] == 192) {
    // insertion order: sa1,sa2,sa3 (W,b,gamma,beta each), fc1, fc2, fc3(W,b)
    int i = 1;
    for (int l = 0; l < 11; ++l) {
      Wl[l]  = (const float*)d_in[i++];
      bl[l]  = (const float*)d_in[i++];
      gl[l]  = (const float*)d_in[i++];
      bel[l] = (const float*)d_in[i++];
    }
    fc3W = (const float*)d_in[i++];
    fc3b = (const float*)d_in[i++];
  } else {
    // jax sorted-key order: fc1, fc2, fc3, sa1, sa2, sa3; per layer W,b,beta,gamma
    int i = 1;
    auto grab = [&](int l) {
      Wl[l]  = (const float*)d_in[i++];
      bl[l]  = (const float*)d_in[i++];
      bel[l] = (const float*)d_in[i++];   // 'beta' sorts before 'gamma'
      gl[l]  = (const float*)d_in[i++];
    };
    grab(9); grab(10);
    fc3W = (const float*)d_in[i++];
    fc3b = (const float*)d_in[i++];
    for (int l = 0; l < 9; ++l) grab(l);
  }

  float* out = (float*)d_out_v;
  float* ws  = (float*)d_ws;
  size_t off = 0;
  auto alloc = [&](size_t nf) { float* p = ws + off; off += nf; return p; };
  float* stats  = alloc((size_t)11 * 2048);
  float* l1_xyz = alloc((size_t)B_ * S1_ * 3);
  float* l2_xyz = alloc((size_t)B_ * S2_ * 3);
  float* l1_pts = alloc((size_t)B_ * S1_ * 128);
  float* l2_pts = alloc((size_t)B_ * S2_ * 128);
  float* wp0    = alloc((size_t)64 * 4);           // sa1 L0 weights, Cin 3->4
  float* wp3    = alloc((size_t)64 * 132);         // sa2 L0 weights, Cin 131->132
  float* wp6    = alloc((size_t)256 * 132);        // sa3 L0 weights, Cin 131->132
  float* bufX   = alloc((size_t)M2_ * 132);        // >= M1_*4 and M3_*132
  float* bufA   = alloc((size_t)M1_ * 64);
  float* bufB   = alloc((size_t)M1_ * 64);
  float* logits = alloc((size_t)B_ * 40);
  float* l3     = out + (size_t)B_ * 40;           // l3_points region of d_out

  hipMemsetAsync(stats, 0, (size_t)11 * 2048 * sizeof(float), stream);
  hipMemsetAsync(l1_pts, 0, (size_t)B_ * S1_ * 128 * sizeof(float), stream);
  hipMemsetAsync(l2_pts, 0, (size_t)B_ * S2_ * 128 * sizeof(float), stream);
  hipMemsetAsync(out, 0, (size_t)out_size * sizeof(float), stream);

  // Pad the three ragged-K weight matrices (Cin 3 -> 4, 131 -> 132).
  pack_w_kernel<<<(64 * 4 + 255) / 256, 256, 0, stream>>>(Wl[0], wp0, 64, 3, 4);
  pack_w_kernel<<<(64 * 132 + 255) / 256, 256, 0, stream>>>(Wl[3], wp3, 64, 131, 132);
  pack_w_kernel<<<(256 * 132 + 255) / 256, 256, 0, stream>>>(Wl[6], wp6, 256, 131, 132);

  // mode: 0 = store+stats, 1 = stats only, 3 = normalize+relu+pool.
  auto gemm = [&](int mode, const float* X, const float* W, int l, float* Y,
                  float* pooled, int M, int Cin, int Cout, int nsample) {
    const int tilesN = Cout >> 4;
    const int nt = (tilesN % 4 == 0) ? 4 : 2;        // all Couts are even in tiles
    const int tiles = (M >> 4) * (tilesN / nt);
    const int grid = (tiles + 3) / 4;                // 4 waves (128 thr) per block
    float* st = stats + (size_t)l * 2048;
    if (mode == 0) {
      if (nt == 4) gemm16<0, 4><<<grid, 128, 0, stream>>>(X, W, bl[l], Y, st, nullptr, nullptr, nullptr, M, Cin, Cout, nsample);
      else         gemm16<0, 2><<<grid, 128, 0, stream>>>(X, W, bl[l], Y, st, nullptr, nullptr, nullptr, M, Cin, Cout, nsample);
    } else if (mode == 1) {
      if (nt == 4) gemm16<1, 4><<<grid, 128, 0, stream>>>(X, W, bl[l], nullptr, st, nullptr, nullptr, nullptr, M, Cin, Cout, nsample);
      else         gemm16<1, 2><<<grid, 128, 0, stream>>>(X, W, bl[l], nullptr, st, nullptr, nullptr, nullptr, M, Cin, Cout, nsample);
    } else {
      if (nt == 4) gemm16<3, 4><<<grid, 128, 0, stream>>>(X, W, bl[l], nullptr, st, gl[l], bel[l], pooled, M, Cin, Cout, nsample);
      else         gemm16<3, 2><<<grid, 128, 0, stream>>>(X, W, bl[l], nullptr, st, gl[l], bel[l], pooled, M, Cin, Cout, nsample);
    }
  };
  auto bn = [&](float* Y, int l, int M, int C) {
    const size_t tot = (size_t)M * C;
    bn_relu_kernel<<<(int)((tot + 255) / 256), 256, 0, stream>>>(
        Y, stats + (size_t)l * 2048, gl[l], bel[l], M, C);
  };

  // ---- stage 1: sample 54, group 28, MLP 3->64->64->128, max-pool ----
  fps_kernel<<<B_, 256, N_ * 3 * sizeof(float), stream>>>(data, N_, S1_, l1_xyz);
  ball_group_kernel<<<(B_ * S1_ * 32 + 127) / 128, 128, 0, stream>>>(
      data, l1_xyz, nullptr, bufX, N_, S1_, K1_, 0, 4, 0.04f);
  gemm(0, bufX, wp0, 0, bufA, nullptr, M1_, 4, 64, 0);    bn(bufA, 0, M1_, 64);
  gemm(0, bufA, Wl[1], 1, bufB, nullptr, M1_, 64, 64, 0); bn(bufB, 1, M1_, 64);
  gemm(1, bufB, Wl[2], 2, nullptr, nullptr, M1_, 64, 128, 0);
  gemm(3, bufB, Wl[2], 2, nullptr, l1_pts, M1_, 64, 128, K1_);

  // ---- stage 2: sample 22, group 8, MLP 131->64->64->128, max-pool ----
  fps_kernel<<<B_, 256, S1_ * 3 * sizeof(float), stream>>>(l1_xyz, S1_, S2_, l2_xyz);
  ball_group_kernel<<<(B_ * S2_ * 32 + 127) / 128, 128, 0, stream>>>(
      l1_xyz, l2_xyz, l1_pts, bufX, S1_, S2_, K2_, 128, 132, 0.04f);
  gemm(0, bufX, wp3, 3, bufA, nullptr, M2_, 132, 64, 0);  bn(bufA, 3, M2_, 64);
  gemm(0, bufA, Wl[4], 4, bufB, nullptr, M2_, 64, 64, 0); bn(bufB, 4, M2_, 64);
  gemm(1, bufB, Wl[5], 5, nullptr, nullptr, M2_, 64, 128, 0);
  gemm(3, bufB, Wl[5], 5, nullptr, l2_pts, M2_, 64, 128, K2_);

  // ---- stage 3: global MLP 131->256->512->1024, max over 22 -> l3_points ----
  concat_x3_kernel<<<(int)(((size_t)M3_ * 132 + 255) / 256), 256, 0, stream>>>(
      l2_xyz, l2_pts, bufX, M3_);
  gemm(0, bufX, wp6, 6, bufA, nullptr, M3_, 132, 256, 0);  bn(bufA, 6, M3_, 256);
  gemm(0, bufA, Wl[7], 7, bufB, nullptr, M3_, 256, 512, 0); bn(bufB, 7, M3_, 512);
  gemm(1, bufB, Wl[8], 8, nullptr, nullptr, M3_, 512, 1024, 0);
  gemm(3, bufB, Wl[8], 8, nullptr, l3, M3_, 512, 1024, S2_);

  // ---- head: fc1, fc2 (batch-norm over B), fc3 + log_softmax ----
  gemm(0, l3, Wl[9], 9, bufA, nullptr, B_, 1024, 576, 0);    bn(bufA, 9, B_, 576);
  gemm(0, bufA, Wl[10], 10, bufB, nullptr, B_, 576, 160, 0); bn(bufB, 10, B_, 160);
  fc3_kernel<<<(B_ * 40 + 255) / 256, 256, 0, stream>>>(bufB, fc3W, fc3b, logits);
  lsm_kernel<<<(B_ * 32 + 255) / 256, 256, 0, stream>>>(logits, out);
}